// UVNetGraphEncoder_9732395893098
// MI455X (gfx1250) — compile-verified
//
#include <hip/hip_runtime.h>
#include <math.h>
#include <stdint.h>

typedef __attribute__((ext_vector_type(2))) float v2f;
typedef __attribute__((ext_vector_type(8))) float v8f;
typedef __attribute__((ext_vector_type(4))) unsigned int v4u;
typedef __attribute__((ext_vector_type(4))) int v4i;
typedef __attribute__((ext_vector_type(8))) int v8i;

#define N_NODES 100000
#define N_EDGES 1600000
#define N_GRAPHS 64
#define DH 64
#define DOUT 128
#define BN_EPS 1e-5f

// ---------------------------------------------------------------------------
// z = (1 + eps[layer]) * h      (vectorized float4)
// ---------------------------------------------------------------------------
__global__ void init_z_kernel(const float4* __restrict__ h, const float* __restrict__ eps,
                              int layer, float4* __restrict__ z, int total4) {
    int i = blockIdx.x * blockDim.x + threadIdx.x;
    if (i >= total4) return;
    float e = 1.0f + eps[layer];
    float4 v = h[i];
    v.x *= e; v.y *= e; v.z *= e; v.w *= e;
    z[i] = v;
}

// ---------------------------------------------------------------------------
// z[dst[e]] += h[src[e]]  : 16 threads per edge, float4 per thread,
// f32 atomics (global_atomic_add_f32, no-return) that resolve in L2.
// ---------------------------------------------------------------------------
__global__ void scatter_add_kernel(const float* __restrict__ h, const int* __restrict__ src,
                                   const int* __restrict__ dst, float* __restrict__ z) {
    int t = blockIdx.x * blockDim.x + threadIdx.x;   // < E*16 = 25.6M, fits int
    int e = t >> 4;
    if (e >= N_EDGES) return;
    int q = (t & 15) << 2;
    int s = src[e];
    int d = dst[e];
    float4 v = *(const float4*)(h + (size_t)s * DH + q);
    float* p = z + (size_t)d * DH + q;
    atomicAdd(p + 0, v.x);
    atomicAdd(p + 1, v.y);
    atomicAdd(p + 2, v.z);
    atomicAdd(p + 3, v.w);
}

// ---------------------------------------------------------------------------
// Tensor Data Mover: async 2D tile (up to 64x64 f32) Global -> LDS.
// LDS row stride = 65 DWORDs via TDM padding (pad 1 DWORD every 64 DWORDs),
// rows beyond `rows` (tensor_dim1) are hardware zero-filled (OOB read = 0).
// D# layout per CDNA5 ISA ch.8 (group0: count/lds/global/type,
// group1: data_size|pad | dims | tile dims | strides).
// ---------------------------------------------------------------------------
__device__ __forceinline__ void tdm_load_tile_f32(unsigned lds_byte_off,
                                                  const float* __restrict__ gptr,
                                                  unsigned rows /*tensor_dim1*/,
                                                  unsigned row_stride_elems) {
    unsigned long long ga = (unsigned long long)(uintptr_t)gptr;
    v4u g0;
    g0.x = 1u;                                              // count=1, user desc
    g0.y = lds_byte_off;                                    // lds_addr
    g0.z = (unsigned)(ga & 0xFFFFFFFFull);                  // global_addr lo
    g0.w = (unsigned)((ga >> 32) & 0x01FFFFFFull)           // global_addr hi (57b)
           | 0x80000000u;                                   // type=2 ("image")
    v8i g1;
    g1[0] = (int)((2u << 16)        // data_size = 2 -> 4 bytes
                  | (1u << 20)      // pad_enable
                  | (5u << 22)      // pad_interval code 5 -> every 64 DWORDs
                  | (0u << 25));    // pad_amount  code 0 -> 1 DWORD  (stride 65)
    g1[1] = (int)(64u << 16);                               // tensor_dim0 = 64 (lo16)
    g1[2] = (int)((rows & 0xFFFFu) << 16);                  // td0 hi=0 | tensor_dim1 lo16
    g1[3] = (int)(((rows >> 16) & 0xFFFFu) | (64u << 16));  // td1 hi16 | tile_dim0 = 64
    g1[4] = 64;                                             // tile_dim1 = 64, tile_dim2 = 0
    g1[5] = (int)row_stride_elems;                          // tensor_dim0_stride lo32
    g1[6] = 0;                                              // stride hi | td1_stride lo
    g1[7] = 0;
    v4i z4 = {0, 0, 0, 0};
#if __clang_major__ >= 23
    v8i z8 = {0, 0, 0, 0, 0, 0, 0, 0};
    __builtin_amdgcn_tensor_load_to_lds(g0, g1, z4, z4, z8, 0);
#else
    __builtin_amdgcn_tensor_load_to_lds(g0, g1, z4, z4, 0);
#endif
}

// ---------------------------------------------------------------------------
// fp32 WMMA helper: one wave computes a 16x64 strip:  acc[nt] = Astrip @ B
// sA: base of this wave's 16-row strip, row stride 65 (padded LDS)
// sB: 64x64 matrix, row stride 65
// Uses V_WMMA_F32_16X16X4_F32; A-frags hoisted and reused across n-tiles.
// ---------------------------------------------------------------------------
__device__ __forceinline__ void wmma_strip_64(const float* sA, const float* sB,
                                              int lane, v8f acc[4]) {
    const int m  = lane & 15;
    const int kb = (lane >> 4) << 1;           // 0 or 2
    v2f a[16];
#pragma unroll
    for (int k = 0; k < 16; ++k) {
        const float* p = sA + m * 65 + (k << 2) + kb;
        a[k].x = p[0];
        a[k].y = p[1];
    }
#pragma unroll
    for (int nt = 0; nt < 4; ++nt) {
        v8f c = {};
#pragma unroll
        for (int k = 0; k < 16; ++k) {
            const float* q = sB + ((k << 2) + kb) * 65 + (nt << 4) + (lane & 15);
            v2f b;
            b.x = q[0];
            b.y = q[65];
            c = __builtin_amdgcn_wmma_f32_16x16x4_f32(false, a[k], false, b,
                                                      (short)0, c, false, false);
        }
        acc[nt] = c;
    }
}

__device__ __forceinline__ float bn_relu(float x, const float* __restrict__ bn, int col) {
    float g = bn[col], b = bn[64 + col], m = bn[128 + col], v = bn[192 + col];
    float s = g * rsqrtf(v + BN_EPS);
    return fmaxf((x - m) * s + b, 0.0f);
}

// ---------------------------------------------------------------------------
// Fused GIN MLP: h_out = relu(bn_o(relu(bn_a(relu(bn1(z@W1+b1))@W2+b2))))
// 128 threads = 4 waves, 64 nodes per block. Each wave owns a 16-row strip.
// Input tile + both weight matrices staged by the Tensor Data Mover.
// ---------------------------------------------------------------------------
__global__ __launch_bounds__(128)
void gin_mlp_kernel(const float* __restrict__ z,
                    const float* __restrict__ W1, const float* __restrict__ b1,
                    const float* __restrict__ bn1,
                    const float* __restrict__ W2, const float* __restrict__ b2,
                    const float* __restrict__ abn, const float* __restrict__ obn,
                    float* __restrict__ out) {
    __shared__ float sZ[64 * 65];
    __shared__ float sW1[64 * 65];
    __shared__ float sW2[64 * 65];

    const int tid  = threadIdx.x;
    const int lane = tid & 31;
    const int wave = tid >> 5;
    const int row0 = blockIdx.x * 64;

    // Wave 0 issues three async TDM tile loads, then drains TENSORcnt.
    if (tid < 32) {
        unsigned rows = (unsigned)(N_NODES - row0);   // ragged tail -> HW zero-fill
        tdm_load_tile_f32((unsigned)(uintptr_t)(void*)sZ,
                          z + (size_t)row0 * DH, rows, DH);
        tdm_load_tile_f32((unsigned)(uintptr_t)(void*)sW1, W1, 64u, DH);
        tdm_load_tile_f32((unsigned)(uintptr_t)(void*)sW2, W2, 64u, DH);
        __builtin_amdgcn_s_wait_tensorcnt(0);
    }
    __syncthreads();

    const float* strip = sZ + wave * 16 * 65;

    // ---- GEMM 1: t = relu(bn1(z @ W1 + b1)) ----
    v8f acc[4];
    wmma_strip_64(strip, sW1, lane, acc);

    // write t back into this wave's own strip (no cross-wave LDS hazard)
    float* wstrip = sZ + wave * 16 * 65;
#pragma unroll
    for (int nt = 0; nt < 4; ++nt) {
        int col = (nt << 4) + (lane & 15);
#pragma unroll
        for (int v = 0; v < 8; ++v) {
            int r = v + ((lane >> 4) << 3);       // 0..15 within strip
            float val = acc[nt][v] + b1[col];
            wstrip[r * 65 + col] = bn_relu(val, bn1, col);
        }
    }
    __syncthreads();

    // ---- GEMM 2: h = relu(bn_o(relu(bn_a(t @ W2 + b2)))) ----
    v8f acc2[4];
    wmma_strip_64(strip, sW2, lane, acc2);

#pragma unroll
    for (int nt = 0; nt < 4; ++nt) {
        int col = (nt << 4) + (lane & 15);
#pragma unroll
        for (int v = 0; v < 8; ++v) {
            int r = v + ((lane >> 4) << 3);
            int grow = row0 + wave * 16 + r;
            float val = acc2[nt][v] + b2[col];
            val = bn_relu(val, abn, col);
            val = bn_relu(val, obn, col);
            if (grow < N_NODES) out[(size_t)grow * DH + col] = val;
        }
    }
}

// ---------------------------------------------------------------------------
// Per-graph max pool (gid sorted) — deterministic, binary-searched bounds.
// One block per graph, 256 threads = 4 row-slices x 64 dims.
// ---------------------------------------------------------------------------
__device__ __forceinline__ int lower_bound_i(const int* __restrict__ a, int n, int key) {
    int lo = 0, hi = n;
    while (lo < hi) {
        int mid = (lo + hi) >> 1;
        if (a[mid] < key) lo = mid + 1; else hi = mid;
    }
    return lo;
}

__global__ void pool_max_kernel(const float* __restrict__ hrep, const int* __restrict__ gid,
                                float* __restrict__ pooled) {
    __shared__ float red[4][64];
    int g  = blockIdx.x;
    int lo = lower_bound_i(gid, N_NODES, g);
    int hi = lower_bound_i(gid, N_NODES, g + 1);
    int d  = threadIdx.x & 63;
    int s  = threadIdx.x >> 6;
    float mx = -INFINITY;
    for (int r = lo + s; r < hi; r += 4)
        mx = fmaxf(mx, hrep[(size_t)r * DH + d]);
    red[s][d] = mx;
    __syncthreads();
    if (threadIdx.x < 64) {
        float m = fmaxf(fmaxf(red[0][d], red[1][d]), fmaxf(red[2][d], red[3][d]));
        pooled[g * DH + d] = m;
    }
}

// ---------------------------------------------------------------------------
// score[64,128] = sum_i pooled[i] @ pred_W[i] + sum_i pred_b[i]
// One 256-thread block, 8 waves, fp32 WMMA, K accumulated over 3 layers.
// ---------------------------------------------------------------------------
__global__ __launch_bounds__(256)
void pred_kernel(const float* __restrict__ pooled,   // [3][64][64]
                 const float* __restrict__ pW,       // [3][64][128]
                 const float* __restrict__ pb,       // [3][128]
                 float* __restrict__ score) {        // [64][128]
    const int lane  = threadIdx.x & 31;
    const int wave  = threadIdx.x >> 5;
    const int mt    = wave >> 1;             // row tile 0..3
    const int nbase = (wave & 1) * 64;       // column half
    const int m     = lane & 15;
    const int kb    = (lane >> 4) << 1;

#pragma unroll
    for (int nt = 0; nt < 4; ++nt) {
        int col = nbase + (nt << 4) + (lane & 15);
        v8f c = {};
        for (int i = 0; i < 3; ++i) {
            const float* A = pooled + i * 64 * 64 + ((mt << 4) + m) * 64;
            const float* B = pW + i * 64 * 128;
#pragma unroll
            for (int k = 0; k < 16; ++k) {
                v2f a, b;
                a.x = A[(k << 2) + kb];
                a.y = A[(k << 2) + kb + 1];
                b.x = B[((k << 2) + kb) * 128 + col];
                b.y = B[((k << 2) + kb + 1) * 128 + col];
                c = __builtin_amdgcn_wmma_f32_16x16x4_f32(false, a, false, b,
                                                          (short)0, c, false, false);
            }
        }
        float bias = pb[col] + pb[128 + col] + pb[256 + col];
#pragma unroll
        for (int v = 0; v < 8; ++v) {
            int row = (mt << 4) + v + ((lane >> 4) << 3);
            score[row * 128 + col] = c[v] + bias;
        }
    }
}

// ---------------------------------------------------------------------------
extern "C" void kernel_launch(void* const* d_in, const int* in_sizes, int n_in,
                              void* d_out, int out_size, void* d_ws, size_t ws_size,
                              hipStream_t stream) {
    const float* h   = (const float*)d_in[0];
    const int*   src = (const int*)d_in[1];
    const int*   dst = (const int*)d_in[2];
    const int*   gid = (const int*)d_in[3];
    const float* eps = (const float*)d_in[4];
    const float* W1  = (const float*)d_in[5];
    const float* b1  = (const float*)d_in[6];
    const float* bn1 = (const float*)d_in[7];
    const float* W2  = (const float*)d_in[8];
    const float* b2  = (const float*)d_in[9];
    const float* abn = (const float*)d_in[10];
    const float* obn = (const float*)d_in[11];
    const float* pW  = (const float*)d_in[12];
    const float* pb  = (const float*)d_in[13];

    float* out    = (float*)d_out;                       // [N, 64]
    float* score  = out + (size_t)N_NODES * DH;          // [64, 128]
    float* zbuf   = (float*)d_ws;                        // [N, 64]  (25.6 MB)
    float* pooled = zbuf + (size_t)N_NODES * DH;         // [3, 64, 64]

    const int total4    = N_NODES * DH / 4;              // 1.6M float4
    const int initBlks  = (total4 + 255) / 256;
    const int scatBlks  = (N_EDGES * 16 + 255) / 256;    // 100000
    const int mlpBlks   = (N_NODES + 63) / 64;           // 1563

    // ---- layer 0 ----
    init_z_kernel<<<initBlks, 256, 0, stream>>>((const float4*)h, eps, 0,
                                                (float4*)zbuf, total4);
    scatter_add_kernel<<<scatBlks, 256, 0, stream>>>(h, src, dst, zbuf);
    gin_mlp_kernel<<<mlpBlks, 128, 0, stream>>>(zbuf, W1, b1, bn1, W2, b2, abn, obn, out);
    pool_max_kernel<<<N_GRAPHS, 256, 0, stream>>>(out, gid, pooled + 1 * 64 * 64);

    // ---- layer 1 (h1 currently lives in d_out) ----
    init_z_kernel<<<initBlks, 256, 0, stream>>>((const float4*)out, eps, 1,
                                                (float4*)zbuf, total4);
    scatter_add_kernel<<<scatBlks, 256, 0, stream>>>(out, src, dst, zbuf);
    gin_mlp_kernel<<<mlpBlks, 128, 0, stream>>>(zbuf, W1 + 64 * 64, b1 + 64, bn1 + 256,
                                                W2 + 64 * 64, b2 + 64, abn + 256, obn + 256,
                                                out);
    pool_max_kernel<<<N_GRAPHS, 256, 0, stream>>>(out, gid, pooled + 2 * 64 * 64);
    pool_max_kernel<<<N_GRAPHS, 256, 0, stream>>>(h,   gid, pooled + 0 * 64 * 64);

    // ---- prediction heads ----
    pred_kernel<<<1, 256, 0, stream>>>(pooled, pW, pb, score);
}